// GraphAppnp_81192061764219
// MI455X (gfx1250) — compile-verified
//
#include <hip/hip_runtime.h>

// GraphAppnp sum-aggregation + APPNP alpha blend, CDNA5 (gfx1250).
// Memory-bound (~1.27 GB/call): TDM (tensor_load_to_lds) double-buffered
// staging of neighbor_agg + NT streaming of neighbor / nbr_out.

typedef float f4 __attribute__((ext_vector_type(4)));
typedef int   i4 __attribute__((ext_vector_type(4)));
typedef int   i8 __attribute__((ext_vector_type(8)));

#define ALPHA         0.1f
#define ONE_M_A       0.9f
#define KNBR          32
#define DFEAT         64
#define F4_PER_ROW    (DFEAT / 4)                   // 16 float4 per feature row
#define NODES_PER_BLK 16
#define BLOCK_THREADS 256
#define CHUNK         8                             // k-rows staged per step
#define NCHUNK        (KNBR / CHUNK)                // 4
#define CHUNK_ELEMS   (CHUNK * DFEAT)               // 512 floats per node-chunk
#define NODE_STRIDE   (KNBR * DFEAT)                // 2048 floats between nodes
#define STAGE_F4      (NODES_PER_BLK * CHUNK * F4_PER_ROW)  // 2048 f4 = 32KB

// Low 32 bits of a generic pointer to __shared__ = wave-relative LDS byte addr.
__device__ __forceinline__ unsigned lds_addr_of(const void* p) {
    return (unsigned)(unsigned long long)p;
}

// One TDM descriptor: 2D tile, dim0 = CHUNK_ELEMS contiguous floats (8 k-rows
// of one node), dim1 = NODES_PER_BLK nodes at NODE_STRIDE, linear into LDS.
// Encoding per CDNA5 ISA (groups 2/3 NULL for 2D).
__device__ __forceinline__ void tdm_issue(unsigned lds_byte, const float* gptr,
                                          int rem_nodes) {
    const unsigned long long ga = (unsigned long long)gptr;
    i4 g0;
    g0[0] = 1;                                      // count=1, user descriptor
    g0[1] = (int)lds_byte;                          // lds_addr (bytes)
    g0[2] = (int)(unsigned)(ga & 0xffffffffu);      // global_addr[31:0]
    g0[3] = (int)(((unsigned)(ga >> 32) & 0x01ffffffu) | 0x80000000u);
                                                    // global_addr[56:32] | type=2
    i8 g1;
    g1[0] = 0x00020000;                             // wg_mask=0, data_size=4B
    g1[1] = (CHUNK_ELEMS & 0xffff) << 16;           // tensor_dim0[15:0] @ [63:48]
    g1[2] = (rem_nodes & 0xffff) << 16;             // dim0 hi=0 | tensor_dim1 lo
    g1[3] = (CHUNK_ELEMS << 16);                    // dim1 hi=0 | tile_dim0=512
    g1[4] = NODES_PER_BLK;                          // tile_dim1=16, tile_dim2=0
    g1[5] = NODE_STRIDE;                            // tensor_dim0_stride lo32
    g1[6] = 0;                                      // stride0 hi | stride1 lo
    g1[7] = 0;                                      // stride1 hi
    asm volatile("tensor_load_to_lds %0, %1" :: "s"(g0), "s"(g1) : "memory");
}

__global__ __launch_bounds__(BLOCK_THREADS)
void appnp_agg_kernel(const float* __restrict__ x,
                      const float* __restrict__ agg,      // neighbor_agg
                      const float* __restrict__ h,
                      const float* __restrict__ nbr,      // neighbor
                      float* __restrict__ x_out,
                      float* __restrict__ nbr_out,
                      int N) {
    __shared__ f4 buf[2][STAGE_F4];                 // 64 KB double buffer

    const int tid  = threadIdx.x;
    const int sub  = tid >> 4;                      // node slot in block (0..15)
    const int lane = tid & 15;                      // float4 slot in row (0..15)
    const int nb0  = blockIdx.x * NODES_PER_BLK;
    const int n    = nb0 + sub;
    const bool live   = (n < N);
    const bool issuer = (tid < 32);                 // wave 0 owns TDM + TENSORcnt

    int rem = N - nb0;
    rem = (rem < NODES_PER_BLK) ? rem : NODES_PER_BLK;  // HW zero-fills OOB rows

    const f4* xv   = (const f4*)x;
    const f4* hv   = (const f4*)h;
    const f4* nbrv = (const f4*)nbr;
    f4* xov = (f4*)x_out;
    f4* nov = (f4*)nbr_out;

    const float* gchunk0 = agg + (size_t)nb0 * NODE_STRIDE;

    // x / h rows: small, reused (K+1 uses) -> regular cacheable loads.
    f4 x4 = {0.f, 0.f, 0.f, 0.f}, h4 = {0.f, 0.f, 0.f, 0.f};
    if (live) {
        x4 = xv[(unsigned)n * F4_PER_ROW + lane];
        h4 = hv[(unsigned)n * F4_PER_ROW + lane];
    }

    f4 acc = {0.f, 0.f, 0.f, 0.f};

    if (issuer) tdm_issue(lds_addr_of(&buf[0][0]), gchunk0, rem);

    for (int c = 0; c < NCHUNK; ++c) {
        const int nc = c + 1;
        if (issuer) {
            if (nc < NCHUNK) {
                tdm_issue(lds_addr_of(&buf[nc & 1][0]),
                          gchunk0 + (size_t)nc * CHUNK_ELEMS, rem);
                __builtin_amdgcn_s_wait_tensorcnt(1);   // chunk c landed
            } else {
                __builtin_amdgcn_s_wait_tensorcnt(0);
            }
        }
        __syncthreads();                             // chunk c visible to all

        if (live) {
            const unsigned rowbase = ((unsigned)n * KNBR + (unsigned)c * CHUNK)
                                     * F4_PER_ROW + lane;
            #pragma unroll
            for (int r = 0; r < CHUNK; ++r) {
                const f4 a = buf[c & 1][sub * (CHUNK * F4_PER_ROW)
                                        + r * F4_PER_ROW + lane];
                acc += a;                            // per-node K-sum piece
                const unsigned idx = rowbase + (unsigned)r * F4_PER_ROW;
                // 410MB streams: non-temporal, keep L2 for x/h
                const f4 nb4 = __builtin_nontemporal_load(&nbrv[idx]);
                const f4 o = ONE_M_A * (a + x4) + ALPHA * nb4;
                __builtin_nontemporal_store(o, &nov[idx]);
            }
        }
        __syncthreads();                             // safe to overwrite buf[c&1]
    }

    if (live) {
        const f4 o = ONE_M_A * (x4 + acc) + ALPHA * h4;
        xov[(unsigned)n * F4_PER_ROW + lane] = o;
    }
}

extern "C" void kernel_launch(void* const* d_in, const int* in_sizes, int n_in,
                              void* d_out, int out_size, void* d_ws, size_t ws_size,
                              hipStream_t stream) {
    (void)n_in; (void)out_size; (void)d_ws; (void)ws_size;
    const float* x   = (const float*)d_in[0];   // [N, D]
    const float* agg = (const float*)d_in[1];   // [N, K, D]
    const float* h   = (const float*)d_in[2];   // [N, D]
    const float* nbr = (const float*)d_in[3];   // [N, K, D]

    const int N = in_sizes[0] / DFEAT;          // 50000

    float* x_out   = (float*)d_out;                         // first N*D floats
    float* nbr_out = x_out + (size_t)N * DFEAT;             // then N*K*D floats

    const int blocks = (N + NODES_PER_BLK - 1) / NODES_PER_BLK;  // 3125
    appnp_agg_kernel<<<blocks, BLOCK_THREADS, 0, stream>>>(
        x, agg, h, nbr, x_out, nbr_out, N);
}